// GCN_46145128628865
// MI455X (gfx1250) — compile-verified
//
#include <hip/hip_runtime.h>

typedef __attribute__((ext_vector_type(2))) float v2f;
typedef __attribute__((ext_vector_type(8))) float v8f;

#define F_IN 5
#define HID 16

// ---------------------------------------------------------------------------
// 1) deg[i] = 1.0 (self loop contribution to degree)
// ---------------------------------------------------------------------------
__global__ void k_init_deg(float* __restrict__ deg, int N) {
  int i = blockIdx.x * blockDim.x + threadIdx.x;
  if (i < N) deg[i] = 1.0f;
}

// ---------------------------------------------------------------------------
// 2) deg[dst] += 1 for every edge
// ---------------------------------------------------------------------------
__global__ void k_degree(const int* __restrict__ ei, float* __restrict__ deg, int E) {
  int e = blockIdx.x * blockDim.x + threadIdx.x;
  if (e < E) {
    __builtin_prefetch(&ei[(long)E + e + 16384], 0, 0);
    atomicAdd(&deg[ei[(long)E + e]], 1.0f);
  }
}

// ---------------------------------------------------------------------------
// 3) dinv[i] = rsqrt(deg[i])   (deg >= 1 always, matches jnp.where(deg>0,...))
// ---------------------------------------------------------------------------
__global__ void k_rsqrt(float* __restrict__ d, int N) {
  int i = blockIdx.x * blockDim.x + threadIdx.x;
  if (i < N) d[i] = rsqrtf(d[i]);
}

// ---------------------------------------------------------------------------
// 4) xw = x @ W1  via V_WMMA_F32_16X16X4_F32 (exact fp32 matrix path).
//    One wave -> 16 nodes. K=5 padded to 8 -> two K=4 WMMA steps.
//    Also writes self-loop term: out1 = dinv^2 * xw.
//
//    A layout (16x4 f32): lanes 0-15 hold M=0..15, VGPR0=K0/VGPR1=K1;
//                         lanes16-31 hold M=0..15, VGPR0=K2/VGPR1=K3.
//    B layout (4x16 f32): row striped across lanes within a VGPR (mirror of A).
//    C/D layout (16x16 f32): VGPR r: lanes0-15 -> M=r, lanes16-31 -> M=8+r; N=lane%16.
// ---------------------------------------------------------------------------
__global__ void k_xw_wmma(const float* __restrict__ x,    // [N,5]
                          const float* __restrict__ W1,   // [5,16]
                          const float* __restrict__ dinv, // [N]
                          float* __restrict__ xw,         // [N,16]
                          float* __restrict__ out1,       // [N,16]
                          int N) {
  const int lane = threadIdx.x & 31;
  const int wave = threadIdx.x >> 5;
  const int wavesPerBlock = blockDim.x >> 5;
  const int base = (blockIdx.x * wavesPerBlock + wave) * 16;
  if (base >= N) return;                 // wave-uniform: EXEC stays all-1s for WMMA

  const int half = lane >> 4;            // 0: K0..1 / M rows 0-7 of D; 1: K2..3 / rows 8-15
  const int l    = lane & 15;            // column / row-within-half index
  int row = base + l;
  if (row >= N) row = N - 1;             // lane-arithmetic clamp (no EXEC change)

  const float* xr = x + (long)row * F_IN;
  const int klo = half * 2;

  // K-step 0 covers K = 0..3 of padded input
  v2f a0; a0.x = xr[klo];            a0.y = xr[klo + 1];
  v2f b0; b0.x = W1[klo * HID + l];  b0.y = W1[(klo + 1) * HID + l];

  // K-step 1 covers K = 4..7 (only K=4 is real; rest zero-padded)
  float x4 = xr[4];
  float w4 = W1[4 * HID + l];
  v2f a1; a1.x = half ? 0.0f : x4;  a1.y = 0.0f;
  v2f b1; b1.x = half ? 0.0f : w4;  b1.y = 0.0f;

  v8f c = {};
  c = __builtin_amdgcn_wmma_f32_16x16x4_f32(false, a0, false, b0, (short)0, c, false, false);
  c = __builtin_amdgcn_wmma_f32_16x16x4_f32(false, a1, false, b1, (short)0, c, false, false);

  const long colBase = (long)(base + half * 8) * HID + l;
  if (base + 16 <= N) {
    // Full tile (always the case for N % 16 == 0): straight-line, no EXEC churn.
    float* xp = xw + colBase;
    float* op = out1 + colBase;
    const float* dp = dinv + base + half * 8;
#pragma unroll
    for (int r = 0; r < 8; ++r) {
      const float v = c[r];
      const float di = dp[r];
      xp[(long)r * HID] = v;                  // 16 lanes -> contiguous 64B per row
      op[(long)r * HID] = di * di * v;        // self-loop init of aggregation
    }
  } else {
    // Tail tile (defensive; unused when N is a multiple of 16).
#pragma unroll
    for (int r = 0; r < 8; ++r) {
      const int node = base + half * 8 + r;
      if (node < N) {
        const float v = c[r];
        const float di = dinv[node];
        xw[(long)node * HID + l] = v;
        out1[(long)node * HID + l] = di * di * v;
      }
    }
  }
}

// ---------------------------------------------------------------------------
// 5) Layer-1 aggregation: out1[dst] += dinv[src]*dinv[dst] * xw[src]
//    One thread per edge: one 64B cacheline gather, 16 f32 atomics (L2-resident).
// ---------------------------------------------------------------------------
__global__ void k_agg1(const int* __restrict__ ei, const float* __restrict__ dinv,
                       const float* __restrict__ xw, float* __restrict__ out1, int E) {
  int e = blockIdx.x * blockDim.x + threadIdx.x;
  if (e >= E) return;
  __builtin_prefetch(&ei[e + 16384], 0, 0);
  __builtin_prefetch(&ei[(long)E + e + 16384], 0, 0);
  const int s = ei[e];
  const int d = ei[(long)E + e];
  const float nrm = dinv[s] * dinv[d];
  const float4* xs = (const float4*)(xw + (long)s * HID);
  float* od = out1 + (long)d * HID;
  float4 v0 = xs[0], v1 = xs[1], v2 = xs[2], v3 = xs[3];
  atomicAdd(od + 0,  nrm * v0.x);  atomicAdd(od + 1,  nrm * v0.y);
  atomicAdd(od + 2,  nrm * v0.z);  atomicAdd(od + 3,  nrm * v0.w);
  atomicAdd(od + 4,  nrm * v1.x);  atomicAdd(od + 5,  nrm * v1.y);
  atomicAdd(od + 6,  nrm * v1.z);  atomicAdd(od + 7,  nrm * v1.w);
  atomicAdd(od + 8,  nrm * v2.x);  atomicAdd(od + 9,  nrm * v2.y);
  atomicAdd(od + 10, nrm * v2.z);  atomicAdd(od + 11, nrm * v2.w);
  atomicAdd(od + 12, nrm * v3.x);  atomicAdd(od + 13, nrm * v3.y);
  atomicAdd(od + 14, nrm * v3.z);  atomicAdd(od + 15, nrm * v3.w);
}

// ---------------------------------------------------------------------------
// 6) h = relu(out1 + b1); z = h @ W2 (16-wide matvec -> scalar dot);
//    out[i] = b2 + dinv[i]^2 * z[i]  (self-loop init of layer-2 aggregation)
// ---------------------------------------------------------------------------
__global__ void k_layer2(const float* __restrict__ out1, const float* __restrict__ b1,
                         const float* __restrict__ W2, const float* __restrict__ b2,
                         const float* __restrict__ dinv, float* __restrict__ z,
                         float* __restrict__ out, int N) {
  int i = blockIdx.x * blockDim.x + threadIdx.x;
  if (i >= N) return;
  const float4* hr = (const float4*)(out1 + (long)i * HID);
  const float4* bb = (const float4*)b1;
  const float4* ww = (const float4*)W2;
  float acc = 0.0f;
#pragma unroll
  for (int q = 0; q < 4; ++q) {
    float4 v = hr[q], b = bb[q], w = ww[q];
    acc = fmaf(fmaxf(v.x + b.x, 0.0f), w.x, acc);
    acc = fmaf(fmaxf(v.y + b.y, 0.0f), w.y, acc);
    acc = fmaf(fmaxf(v.z + b.z, 0.0f), w.z, acc);
    acc = fmaf(fmaxf(v.w + b.w, 0.0f), w.w, acc);
  }
  z[i] = acc;
  const float di = dinv[i];
  out[i] = b2[0] + di * di * acc;
}

// ---------------------------------------------------------------------------
// 7) Layer-2 aggregation: out[dst] += dinv[src]*dinv[dst] * z[src]
// ---------------------------------------------------------------------------
__global__ void k_agg2(const int* __restrict__ ei, const float* __restrict__ dinv,
                       const float* __restrict__ z, float* __restrict__ out, int E) {
  int e = blockIdx.x * blockDim.x + threadIdx.x;
  if (e >= E) return;
  __builtin_prefetch(&ei[e + 16384], 0, 0);
  __builtin_prefetch(&ei[(long)E + e + 16384], 0, 0);
  const int s = ei[e];
  const int d = ei[(long)E + e];
  atomicAdd(&out[d], dinv[s] * dinv[d] * z[s]);
}

// ---------------------------------------------------------------------------
extern "C" void kernel_launch(void* const* d_in, const int* in_sizes, int n_in,
                              void* d_out, int out_size, void* d_ws, size_t ws_size,
                              hipStream_t stream) {
  const float* x  = (const float*)d_in[0];   // [N,5]
  const int*   ei = (const int*)d_in[1];     // [2,E]
  const float* W1 = (const float*)d_in[2];   // [5,16]
  const float* b1 = (const float*)d_in[3];   // [16]
  const float* W2 = (const float*)d_in[4];   // [16,1]
  const float* b2 = (const float*)d_in[5];   // [1]
  float* out = (float*)d_out;                // [N,1]

  const int N = in_sizes[0] / F_IN;
  const int E = in_sizes[1] / 2;

  // Carve workspace (256B-aligned): dinv[N], xw[N*16], out1[N*16], z[N]
  char* ws = (char*)d_ws;
  size_t off = 0;
  auto carve = [&](size_t bytes) -> void* {
    off = (off + 255) & ~(size_t)255;
    void* p = ws + off;
    off += bytes;
    return p;
  };
  float* dinv = (float*)carve((size_t)N * sizeof(float));
  float* xw   = (float*)carve((size_t)N * HID * sizeof(float));
  float* out1 = (float*)carve((size_t)N * HID * sizeof(float));
  float* z    = (float*)carve((size_t)N * sizeof(float));
  (void)ws_size; (void)n_in; (void)out_size;

  const int B = 256;                        // 8 wave32s / block
  const int gN = (N + B - 1) / B;
  const int gE = (E + B - 1) / B;
  const int wavesNeeded   = (N + 15) / 16;  // 16 nodes per wave in the WMMA kernel
  const int wavesPerBlock = B / 32;
  const int gW = (wavesNeeded + wavesPerBlock - 1) / wavesPerBlock;

  hipLaunchKernelGGL(k_init_deg, dim3(gN), dim3(B), 0, stream, dinv, N);
  hipLaunchKernelGGL(k_degree,   dim3(gE), dim3(B), 0, stream, ei, dinv, E);
  hipLaunchKernelGGL(k_rsqrt,    dim3(gN), dim3(B), 0, stream, dinv, N);
  hipLaunchKernelGGL(k_xw_wmma,  dim3(gW), dim3(B), 0, stream, x, W1, dinv, xw, out1, N);
  hipLaunchKernelGGL(k_agg1,     dim3(gE), dim3(B), 0, stream, ei, dinv, xw, out1, E);
  hipLaunchKernelGGL(k_layer2,   dim3(gN), dim3(B), 0, stream, out1, b1, W2, b2, dinv, z, out, N);
  hipLaunchKernelGGL(k_agg2,     dim3(gE), dim3(B), 0, stream, ei, dinv, z, out, E);
}